// QwenVL_PartF_20968030339742
// MI455X (gfx1250) — compile-verified
//
#include <hip/hip_runtime.h>
#include <hip/hip_bf16.h>

// ---------------- constants (from reference) ----------------
#define LNUM 4
#define BSZ  3
#define DMODEL 2048
#define NH   16
#define HDIM 128
#define KVH  2
#define GQA  8          // NH / KVH
#define FDIM 6144
#define VOCAB 32000
#define TCTX 4096
#define TPAST 4095
#define EPSV 1e-6f
#define QK_SCALE 0.29730177875068026f   // 128^-0.25

typedef __attribute__((ext_vector_type(2))) float v2f;
typedef __attribute__((ext_vector_type(8))) float v8f;

// ---------------- small utility kernels ----------------
__global__ void copy_f32(const float* __restrict__ src, float* __restrict__ dst, int n) {
    int i = blockIdx.x * 256 + threadIdx.x;
    if (i < n) dst[i] = src[i];
}

__global__ void add_vec(float* __restrict__ y, const float* __restrict__ a, int n) {
    int i = blockIdx.x * 256 + threadIdx.x;
    if (i < n) y[i] += a[i];
}

__global__ void silu_mul(float* __restrict__ g, const float* __restrict__ u, int n) {
    int i = blockIdx.x * 256 + threadIdx.x;
    if (i < n) { float x = g[i]; g[i] = (x / (1.f + __expf(-x))) * u[i]; }
}

// repack past KV caches into output caches (T_PAST -> T stride change)
__global__ void copy_past(const float* __restrict__ pk, const float* __restrict__ pv,
                          float* __restrict__ ok, float* __restrict__ ov) {
    const long long NK = (long long)LNUM * BSZ * KVH * HDIM * TPAST; // per tensor
    for (long long n = (long long)blockIdx.x * 256 + threadIdx.x; n < NK;
         n += (long long)gridDim.x * 256) {
        // K: [.., HDIM, TPAST] -> [.., HDIM, TCTX]
        long long row = n / TPAST, t = n % TPAST;
        ok[row * TCTX + t] = pk[n];
        // V: [.., TPAST, HDIM] -> [.., TCTX, HDIM] : contiguous per (l,b,kv)
        long long blk = n / ((long long)TPAST * HDIM), off = n % ((long long)TPAST * HDIM);
        ov[blk * ((long long)TCTX * HDIM) + off] = pv[n];
    }
}

// ---------------- rmsnorm over D=2048, B rows ----------------
__global__ void rmsnorm_k(const float* __restrict__ x, const float* __restrict__ w,
                          float* __restrict__ y) {
    int b = blockIdx.x;
    __shared__ float red[256];
    const float* xb = x + (size_t)b * DMODEL;
    float s = 0.f;
    for (int d = threadIdx.x; d < DMODEL; d += 256) { float v = xb[d]; s += v * v; }
    red[threadIdx.x] = s; __syncthreads();
    for (int o = 128; o > 0; o >>= 1) {
        if (threadIdx.x < o) red[threadIdx.x] += red[threadIdx.x + o];
        __syncthreads();
    }
    float r = rsqrtf(red[0] / (float)DMODEL + EPSV);
    for (int d = threadIdx.x; d < DMODEL; d += 256)
        y[(size_t)b * DMODEL + d] = xb[d] * r * w[d];
}

// ---------------- split-K GEMV: y[b,n] = sum_k x[b,k] * W[k,n], B=3 ----------------
// grid (ceil(N/256), KSPLIT), block 256. part layout: [ks][3][N]
__global__ void gemv_partial(const float* __restrict__ x, const float* __restrict__ W,
                             float* __restrict__ part, int K, int N, int Kr) {
    __shared__ float xs[3 * 1024];
    const int ks = blockIdx.y;
    const int k0 = ks * Kr;
    const int col = blockIdx.x * 256 + threadIdx.x;
    float a0 = 0.f, a1 = 0.f, a2 = 0.f;
    for (int c0 = 0; c0 < Kr; c0 += 1024) {
        int cl = Kr - c0; if (cl > 1024) cl = 1024;
        __syncthreads();
        for (int idx = threadIdx.x; idx < 3 * cl; idx += 256) {
            int bb = idx / cl, dd = idx % cl;
            xs[bb * 1024 + dd] = x[(size_t)bb * K + k0 + c0 + dd];
        }
        __syncthreads();
        if (col < N) {
            const float* Wp = W + (size_t)(k0 + c0) * N + col;
            for (int d = 0; d < cl; ++d) {
                float wv = Wp[(size_t)d * N];      // coalesced across threads
                a0 += xs[d] * wv;                  // LDS broadcast (no conflicts)
                a1 += xs[1024 + d] * wv;
                a2 += xs[2048 + d] * wv;
            }
        }
    }
    if (col < N) {
        part[((size_t)ks * 3 + 0) * N + col] = a0;
        part[((size_t)ks * 3 + 1) * N + col] = a1;
        part[((size_t)ks * 3 + 2) * N + col] = a2;
    }
}

// y[b,n] = (res? res : 0) + sum_ks part ; grid ceil(3N/256)
__global__ void gemv_combine(const float* __restrict__ part, const float* __restrict__ res,
                             float* __restrict__ y, int N, int ksplit) {
    int i = blockIdx.x * 256 + threadIdx.x;
    if (i >= 3 * N) return;
    int b = i / N, n = i % N;
    float s = res ? res[i] : 0.f;
    for (int ks = 0; ks < ksplit; ++ks) s += part[((size_t)ks * 3 + b) * N + n];
    y[i] = s;
}

// ---------------- per-head q/k RMSNorm + RoPE; writes new K col / V row ----------------
// grid: B*(NH+KVH) blocks of 128 threads
__global__ void qk_rope(const float* __restrict__ qraw, const float* __restrict__ kraw,
                        const float* __restrict__ vraw,
                        const float* __restrict__ wqn, const float* __restrict__ wkn,
                        const float* __restrict__ cosv, const float* __restrict__ sinv,
                        float* __restrict__ qout, float* __restrict__ okL,
                        float* __restrict__ ovL) {
    int blk = blockIdx.x;
    int b = blk / (NH + KVH);
    int r = blk % (NH + KVH);
    int d = threadIdx.x;
    __shared__ float xn[HDIM];
    __shared__ float red[HDIM];
    if (r < NH) {
        float v = qraw[((size_t)b * NH + r) * HDIM + d];
        red[d] = v * v; __syncthreads();
        for (int o = 64; o > 0; o >>= 1) { if (d < o) red[d] += red[d + o]; __syncthreads(); }
        float rr = rsqrtf(red[0] / (float)HDIM + EPSV);
        xn[d] = v * rr * wqn[d] * QK_SCALE; __syncthreads();
        float rot = (d < 64) ? -xn[d + 64] : xn[d - 64];
        qout[((size_t)b * NH + r) * HDIM + d] = xn[d] * cosv[d] + rot * sinv[d];
    } else {
        int kv = r - NH;
        float v = kraw[((size_t)b * KVH + kv) * HDIM + d];
        red[d] = v * v; __syncthreads();
        for (int o = 64; o > 0; o >>= 1) { if (d < o) red[d] += red[d + o]; __syncthreads(); }
        float rr = rsqrtf(red[0] / (float)HDIM + EPSV);
        xn[d] = v * rr * wkn[d] * QK_SCALE; __syncthreads();
        float rot = (d < 64) ? -xn[d + 64] : xn[d - 64];
        // K cache layout [B][KVH][HDIM][TCTX]; new column at t = TPAST
        okL[(((size_t)b * KVH + kv) * HDIM + d) * TCTX + TPAST] = xn[d] * cosv[d] + rot * sinv[d];
        // V cache layout [B][KVH][TCTX][HDIM]; new row at t = TPAST
        ovL[((size_t)b * KVH + kv) * TCTX * HDIM + (size_t)TPAST * HDIM + d] =
            vraw[((size_t)b * KVH + kv) * HDIM + d];
    }
}

// ---------------- attention scores via fp32 WMMA: S[8(+8 pad),16-tile of T] ----------------
// grid B*KVH blocks, 256 threads (8 waves); each wave owns a 16-wide t-tile, strided.
__global__ void attn_scores_wmma(const float* __restrict__ q,    // [B][NH][HDIM]
                                 const float* __restrict__ Kc,   // [B][KVH][HDIM][TCTX]
                                 float* __restrict__ scores) {   // [B][NH][TCTX]
    int b = blockIdx.x / KVH, kvh = blockIdx.x % KVH;
    __shared__ float Qs[16][HDIM + 1];   // padded: avoid 16-way bank conflicts
    int tid = threadIdx.x;
    for (int idx = tid; idx < 16 * HDIM; idx += 256) {
        int m = idx >> 7, d = idx & (HDIM - 1);
        Qs[m][d] = (m < GQA) ? q[((size_t)b * NH + kvh * GQA + m) * HDIM + d] : 0.f;
    }
    __syncthreads();
    const float* Kbase = Kc + (size_t)(b * KVH + kvh) * HDIM * TCTX;
    int wave = tid >> 5, lane = tid & 31;
    int n = lane & 15, khalf = lane >> 4;       // A: M=lane%16, K-pair=lane/16; B symmetric
    for (int tile = wave; tile < TCTX / 16; tile += 8) {
        int t0 = tile * 16;
        v8f c = {};
        for (int d = 0; d < HDIM; d += 4) {
            v2f a, bb;
            a.x = Qs[n][d + khalf * 2 + 0];
            a.y = Qs[n][d + khalf * 2 + 1];
            const float* kp = Kbase + (size_t)(d + khalf * 2) * TCTX + t0 + n;
            bb.x = kp[0];
            bb.y = kp[TCTX];
            c = __builtin_amdgcn_wmma_f32_16x16x4_f32(false, a, false, bb,
                                                      (short)0, c, false, false);
        }
        if (lane < 16) {                        // rows 0..7 live in lanes 0-15, VGPR r
            #pragma unroll
            for (int r = 0; r < GQA; ++r)
                scores[((size_t)(b * NH + kvh * GQA + r)) * TCTX + t0 + lane] = c[r];
        }
    }
}

// ---------------- row softmax over T (with reference mask term) ----------------
__global__ void softmax_row(float* __restrict__ sc, const float* __restrict__ ascale) {
    int row = blockIdx.x;                       // b*NH + h
    float* p = sc + (size_t)row * TCTX;
    float asc = ascale[0];
    __shared__ float red[256];
    int tid = threadIdx.x;
    float mx = -3.4e38f;
    for (int t = tid; t < TCTX; t += 256) {
        float v = p[t] + ((t > 0) ? -128.f * asc : 0.f);
        mx = fmaxf(mx, v);
    }
    red[tid] = mx; __syncthreads();
    for (int o = 128; o > 0; o >>= 1) {
        if (tid < o) red[tid] = fmaxf(red[tid], red[tid + o]);
        __syncthreads();
    }
    mx = red[0]; __syncthreads();
    float s = 0.f;
    for (int t = tid; t < TCTX; t += 256) {
        float v = p[t] + ((t > 0) ? -128.f * asc : 0.f);
        float e = __expf(v - mx);
        p[t] = e; s += e;
    }
    red[tid] = s; __syncthreads();
    for (int o = 128; o > 0; o >>= 1) {
        if (tid < o) red[tid] += red[tid + o];
        __syncthreads();
    }
    float inv = 1.f / red[0];
    for (int t = tid; t < TCTX; t += 256) p[t] *= inv;
}

// ---------------- attention P·V via fp32 WMMA ----------------
// grid B*KVH blocks, 256 threads; wave w accumulates d-tile [w*16, w*16+16)
#define AV_CH 512
__global__ void attn_av_wmma(const float* __restrict__ P,    // probs [B][NH][TCTX]
                             const float* __restrict__ Vc,   // [B][KVH][TCTX][HDIM]
                             float* __restrict__ out) {      // [B][NH*HDIM]
    int b = blockIdx.x / KVH, kvh = blockIdx.x % KVH;
    __shared__ float Ps[16][AV_CH + 1];        // padded against bank conflicts
    int tid = threadIdx.x, wave = tid >> 5, lane = tid & 31;
    int n = lane & 15, khalf = lane >> 4;
    int dcol0 = wave * 16;
    const float* Vbase = Vc + (size_t)(b * KVH + kvh) * TCTX * HDIM;
    const float* Pbase = P + (size_t)(b * NH + kvh * GQA) * TCTX;
    v8f c = {};
    for (int c0 = 0; c0 < TCTX; c0 += AV_CH) {
        __syncthreads();
        for (int idx = tid; idx < 16 * AV_CH; idx += 256) {
            int m = idx / AV_CH, t = idx % AV_CH;
            Ps[m][t] = (m < GQA) ? Pbase[(size_t)m * TCTX + c0 + t] : 0.f;
        }
        __syncthreads();
        for (int t = 0; t < AV_CH; t += 4) {
            v2f a, bb;
            a.x = Ps[n][t + khalf * 2 + 0];
            a.y = Ps[n][t + khalf * 2 + 1];
            const float* vp = Vbase + (size_t)(c0 + t + khalf * 2) * HDIM + dcol0 + n;
            bb.x = vp[0];
            bb.y = vp[HDIM];
            c = __builtin_amdgcn_wmma_f32_16x16x4_f32(false, a, false, bb,
                                                      (short)0, c, false, false);
        }
    }
    if (lane < 16) {
        #pragma unroll
        for (int r = 0; r < GQA; ++r)
            out[(size_t)b * (NH * HDIM) + (kvh * GQA + r) * HDIM + dcol0 + lane] = c[r];
    }
}

// ---------------- host orchestration ----------------
extern "C" void kernel_launch(void* const* d_in, const int* in_sizes, int n_in,
                              void* d_out, int out_size, void* d_ws, size_t ws_size,
                              hipStream_t stream) {
    const float* hidden  = (const float*)d_in[0];
    const float* past_k  = (const float*)d_in[1];
    const float* past_v  = (const float*)d_in[2];
    const float* deep    = (const float*)d_in[3];
    const float* cosq    = (const float*)d_in[4];
    const float* sinq    = (const float*)d_in[5];
    // d_in[6], d_in[7] (cos_k/sin_k) hold the identical 128 values -> reuse cosq/sinq
    const float* ascale  = (const float*)d_in[8];
    const float* w_in_ln = (const float*)d_in[9];
    const float* wq      = (const float*)d_in[10];
    const float* wk      = (const float*)d_in[11];
    const float* wv      = (const float*)d_in[12];
    const float* wqn     = (const float*)d_in[13];
    const float* wkn     = (const float*)d_in[14];
    const float* wo      = (const float*)d_in[15];
    const float* w_post  = (const float*)d_in[16];
    const float* wg      = (const float*)d_in[17];
    const float* wu      = (const float*)d_in[18];
    const float* wd      = (const float*)d_in[19];
    const float* wfn     = (const float*)d_in[20];
    const float* wlm     = (const float*)d_in[21];

    float* logits = (float*)d_out;                                  // [3][32000]
    float* okAll  = logits + (size_t)BSZ * VOCAB;                   // [L][B][KVH][HDIM][TCTX]
    float* ovAll  = okAll + (size_t)LNUM * BSZ * KVH * HDIM * TCTX; // [L][B][KVH][TCTX][HDIM]

    float* ws    = (float*)d_ws;
    float* h     = ws;               // [3][2048]
    float* hn    = h     + 6144;     // [3][2048]
    float* qraw  = hn    + 6144;     // [3][2048]
    float* kraw  = qraw  + 6144;     // [3][256]
    float* vraw  = kraw  + 768;      // [3][256]
    float* qrope = vraw  + 768;      // [3][2048]
    float* scor  = qrope + 6144;     // [3][16][4096]
    float* attn  = scor  + 196608;   // [3][2048]
    float* gate  = attn  + 6144;     // [3][6144]
    float* up    = gate  + 18432;    // [3][6144]
    float* part  = up    + 18432;    // [8][3][32000] max

    const int KS = 8;                // split-K factor

    copy_past<<<4096, 256, 0, stream>>>(past_k, past_v, okAll, ovAll);
    copy_f32<<<24, 256, 0, stream>>>(hidden, h, 6144);

    for (int l = 0; l < LNUM; ++l) {
        float* okL = okAll + (size_t)l * BSZ * KVH * HDIM * TCTX;
        float* ovL = ovAll + (size_t)l * BSZ * KVH * HDIM * TCTX;

        rmsnorm_k<<<BSZ, 256, 0, stream>>>(h, w_in_ln + (size_t)l * DMODEL, hn);

        // q/k/v projections
        gemv_partial<<<dim3(8, KS), 256, 0, stream>>>(hn, wq + (size_t)l * DMODEL * 2048,
                                                      part, DMODEL, 2048, DMODEL / KS);
        gemv_combine<<<24, 256, 0, stream>>>(part, nullptr, qraw, 2048, KS);
        gemv_partial<<<dim3(1, KS), 256, 0, stream>>>(hn, wk + (size_t)l * DMODEL * 256,
                                                      part, DMODEL, 256, DMODEL / KS);
        gemv_combine<<<3, 256, 0, stream>>>(part, nullptr, kraw, 256, KS);
        gemv_partial<<<dim3(1, KS), 256, 0, stream>>>(hn, wv + (size_t)l * DMODEL * 256,
                                                      part, DMODEL, 256, DMODEL / KS);
        gemv_combine<<<3, 256, 0, stream>>>(part, nullptr, vraw, 256, KS);

        qk_rope<<<BSZ * (NH + KVH), 128, 0, stream>>>(qraw, kraw, vraw,
                                                      wqn + (size_t)l * HDIM,
                                                      wkn + (size_t)l * HDIM,
                                                      cosq, sinq, qrope, okL, ovL);

        attn_scores_wmma<<<BSZ * KVH, 256, 0, stream>>>(qrope, okL, scor);
        softmax_row<<<BSZ * NH, 256, 0, stream>>>(scor, ascale);
        attn_av_wmma<<<BSZ * KVH, 256, 0, stream>>>(scor, ovL, attn);

        // output projection + residual
        gemv_partial<<<dim3(8, KS), 256, 0, stream>>>(attn, wo + (size_t)l * 2048 * DMODEL,
                                                      part, 2048, DMODEL, 2048 / KS);
        gemv_combine<<<24, 256, 0, stream>>>(part, h, h, DMODEL, KS);

        // MLP
        rmsnorm_k<<<BSZ, 256, 0, stream>>>(h, w_post + (size_t)l * DMODEL, hn);
        gemv_partial<<<dim3(24, KS), 256, 0, stream>>>(hn, wg + (size_t)l * DMODEL * FDIM,
                                                       part, DMODEL, FDIM, DMODEL / KS);
        gemv_combine<<<72, 256, 0, stream>>>(part, nullptr, gate, FDIM, KS);
        gemv_partial<<<dim3(24, KS), 256, 0, stream>>>(hn, wu + (size_t)l * DMODEL * FDIM,
                                                       part, DMODEL, FDIM, DMODEL / KS);
        gemv_combine<<<72, 256, 0, stream>>>(part, nullptr, up, FDIM, KS);
        silu_mul<<<72, 256, 0, stream>>>(gate, up, BSZ * FDIM);
        gemv_partial<<<dim3(8, KS), 256, 0, stream>>>(gate, wd + (size_t)l * FDIM * DMODEL,
                                                      part, FDIM, DMODEL, FDIM / KS);
        gemv_combine<<<24, 256, 0, stream>>>(part, h, h, DMODEL, KS);

        if (l < 3) add_vec<<<24, 256, 0, stream>>>(h, deep + (size_t)l * BSZ * DMODEL,
                                                   BSZ * DMODEL);
    }

    // final norm + LM head
    rmsnorm_k<<<BSZ, 256, 0, stream>>>(h, wfn, hn);
    gemv_partial<<<dim3(125, KS), 256, 0, stream>>>(hn, wlm, part, DMODEL, VOCAB, DMODEL / KS);
    gemv_combine<<<375, 256, 0, stream>>>(part, nullptr, logits, VOCAB, KS);
}